// CG3Model_79791902425582
// MI455X (gfx1250) — compile-verified
//
#include <hip/hip_runtime.h>
#include <hip/hip_bf16.h>

typedef __attribute__((ext_vector_type(2))) float v2f;
typedef __attribute__((ext_vector_type(8))) float v8f;

#define HID 128
#define SA_STRIDE 130   // padded LDS row stride (floats) to avoid bank conflicts

// ---------------- gcn_norm kernels ----------------

__global__ void init_deg_kernel(float* __restrict__ deg, int n) {
    int i = blockIdx.x * blockDim.x + threadIdx.x;
    if (i < n) deg[i] = 1.0f;   // self-loop weight = 1
}

__global__ void accum_deg_kernel(float* __restrict__ deg, const int* __restrict__ dst,
                                 const float* __restrict__ w, int e) {
    int i = blockIdx.x * blockDim.x + threadIdx.x;
    if (i < e) atomicAdd(&deg[dst[i]], w[i]);
}

__global__ void finalize_dinv_kernel(float* __restrict__ deg, int n) {
    int i = blockIdx.x * blockDim.x + threadIdx.x;
    if (i < n) {
        float d = deg[i];
        deg[i] = (d > 0.0f) ? (1.0f / sqrtf(d)) : 0.0f;   // in-place -> dinv
    }
}

__global__ void edge_norm_kernel(float* __restrict__ norm, const float* __restrict__ dinv,
                                 const int* __restrict__ src, const int* __restrict__ dst,
                                 const float* __restrict__ w, int e) {
    int i = blockIdx.x * blockDim.x + threadIdx.x;
    if (i < e) norm[i] = dinv[src[i]] * w[i] * dinv[dst[i]];
}

// ---------------- fp32 WMMA GEMM ----------------
// out[nRows x OUTCOLS] = op(A[nRows x 128]) @ W[128 x OUTCOLS] (+ bias)
// One wave computes one 16x16 output tile via 32x V_WMMA_F32_16X16X4_F32.
// Block = (ceil(OUTCOLS/16) * 32) threads; blockIdx.x = row tile.
// A tile staged in LDS (ReLU fused); B loaded with compile-time immediate offsets
// from a clamped (always-valid) per-lane column pointer; masking is branch-free.

template <int OUTCOLS, bool RELU_IN, bool HAS_BIAS>
__global__ __launch_bounds__(256) void gemm_k128_wmma(
        const float* __restrict__ A, const float* __restrict__ W,
        const float* __restrict__ bias, float* __restrict__ out,
        int nRows) {
    __shared__ float sA[16 * SA_STRIDE];

    const int rowBase = blockIdx.x * 16;

    // Stage 16x128 A tile into LDS (ReLU fused here when requested)
    for (int idx = threadIdx.x; idx < 16 * HID; idx += blockDim.x) {
        int r = idx >> 7;         // 0..15
        int k = idx & (HID - 1);  // 0..127
        int row = rowBase + r;
        float v = (row < nRows) ? A[row * HID + k] : 0.0f;
        if (RELU_IN) v = fmaxf(v, 0.0f);
        sA[r * SA_STRIDE + k] = v;
    }
    __syncthreads();

    const int wave = threadIdx.x >> 5;
    const int lane = threadIdx.x & 31;
    const int half = lane >> 4;    // selects K{0,1} vs K{2,3} per ISA A/B layout
    const int l    = lane & 15;
    const int col  = wave * 16 + l;
    const bool colOk = (OUTCOLS % 16 == 0) ? true : (col < OUTCOLS);
    // Clamped column: always a valid address; value zeroed via select (v_cndmask).
    const int colc = (OUTCOLS % 16 == 0) ? col : (col < OUTCOLS ? col : OUTCOLS - 1);
    const float* __restrict__ Wc = W + colc;          // per-lane column base
    const float* __restrict__ Wh = Wc + 2 * half * OUTCOLS;  // half shifts K by 2 rows
    const int aBase = l * SA_STRIDE + 2 * half;

    v8f acc = {};
    #pragma unroll
    for (int k0 = 0; k0 < HID; k0 += 4) {
        // A 16x4 fp32 layout: lanes 0-15 & 16-31 both hold M=l; VGPR0=K{0,2}, VGPR1=K{1,3}
        v2f a;
        a.x = sA[aBase + k0];
        a.y = sA[aBase + k0 + 1];
        // B 4x16 fp32 layout: VGPR0 = rows K{0 (lanes0-15), 2 (lanes16-31)}, VGPR1 = K{1,3}
        float bx = Wh[k0 * OUTCOLS];            // compile-time immediate offsets
        float by = Wh[(k0 + 1) * OUTCOLS];
        v2f b;
        b.x = colOk ? bx : 0.0f;                // folds away for OUTCOLS multiple of 16
        b.y = colOk ? by : 0.0f;
        acc = __builtin_amdgcn_wmma_f32_16x16x4_f32(
                /*neg_a=*/false, a, /*neg_b=*/false, b,
                /*c_mod=*/(short)0, acc, /*reuse_a=*/false, /*reuse_b=*/false);
    }

    float bv = 0.0f;
    if (HAS_BIAS) bv = colOk ? bias[colc] : 0.0f;

    // C/D layout: VGPR r -> M = r (lanes 0-15) / M = 8+r (lanes 16-31), N = l
    #pragma unroll
    for (int r = 0; r < 8; ++r) {
        int row = rowBase + r + 8 * half;
        if (row < nRows && colOk) out[row * OUTCOLS + col] = acc[r] + bv;
    }
}

// ---------------- scatter: out = b + dinv^2 * xw  (self-loop + bias init) ----------------

__global__ void scatter_init_kernel(float* __restrict__ acc, const float* __restrict__ xw,
                                    const float* __restrict__ dinv, const float* __restrict__ bias,
                                    int n) {
    int t = blockIdx.x * blockDim.x + threadIdx.x;   // over n*32 float4s
    if (t >= n * 32) return;
    int i  = t >> 5;
    int f4 = t & 31;
    float di = dinv[i];
    float s  = di * di;
    float4 x = ((const float4*)xw)[t];
    float4 b = ((const float4*)bias)[f4];
    float4 o;
    o.x = b.x + s * x.x;  o.y = b.y + s * x.y;
    o.z = b.z + s * x.z;  o.w = b.w + s * x.w;
    ((float4*)acc)[t] = o;
}

// ---------------- scatter: acc[dst] += xw[src] * norm  (one thread per edge-feature) --------

__global__ void scatter_edge_kernel(float* __restrict__ acc, const float* __restrict__ xw,
                                    const int* __restrict__ src, const int* __restrict__ dst,
                                    const float* __restrict__ norm, int e) {
    long long t = (long long)blockIdx.x * blockDim.x + threadIdx.x;
    if (t >= (long long)e * HID) return;
    int ei = (int)(t >> 7);
    int f  = (int)(t & (HID - 1));
    int s  = src[ei];
    int d  = dst[ei];
    float w = norm[ei];
    atomicAdd(&acc[d * HID + f], xw[s * HID + f] * w);
}

// ---------------- L2 normalize (one wave32 per node, float4 per lane) ----------------

__global__ void l2norm_kernel(float* __restrict__ out, const float* __restrict__ in, int n) {
    int wave = threadIdx.x >> 5;
    int lane = threadIdx.x & 31;
    int node = blockIdx.x * (blockDim.x >> 5) + wave;
    if (node >= n) return;
    float4 v = ((const float4*)(in + node * HID))[lane];
    float s = v.x * v.x + v.y * v.y + v.z * v.z + v.w * v.w;
    #pragma unroll
    for (int off = 16; off; off >>= 1) s += __shfl_xor(s, off, 32);
    float rn = 1.0f / fmaxf(sqrtf(s), 1e-12f);
    float4 o; o.x = v.x * rn; o.y = v.y * rn; o.z = v.z * rn; o.w = v.w * rn;
    ((float4*)(out + node * HID))[lane] = o;
}

// ---------------- z = l2norm(alpha*z_gcn + (1-alpha)*z_hgcn) ----------------

__global__ void combine_kernel(float* __restrict__ z, const float* __restrict__ za,
                               const float* __restrict__ zb, int n) {
    int wave = threadIdx.x >> 5;
    int lane = threadIdx.x & 31;
    int node = blockIdx.x * (blockDim.x >> 5) + wave;
    if (node >= n) return;
    float4 a = ((const float4*)(za + node * HID))[lane];
    float4 b = ((const float4*)(zb + node * HID))[lane];
    float4 v;
    v.x = 0.6f * a.x + 0.4f * b.x;  v.y = 0.6f * a.y + 0.4f * b.y;
    v.z = 0.6f * a.z + 0.4f * b.z;  v.w = 0.6f * a.w + 0.4f * b.w;
    float s = v.x * v.x + v.y * v.y + v.z * v.z + v.w * v.w;
    #pragma unroll
    for (int off = 16; off; off >>= 1) s += __shfl_xor(s, off, 32);
    float rn = 1.0f / fmaxf(sqrtf(s), 1e-12f);
    float4 o; o.x = v.x * rn; o.y = v.y * rn; o.z = v.z * rn; o.w = v.w * rn;
    ((float4*)(z + node * HID))[lane] = o;
}

// ---------------- launcher ----------------

extern "C" void kernel_launch(void* const* d_in, const int* in_sizes, int n_in,
                              void* d_out, int out_size, void* d_ws, size_t ws_size,
                              hipStream_t stream) {
    const float* x   = (const float*)d_in[0];
    const int*   ei  = (const int*)d_in[1];
    const float* ew  = (const float*)d_in[2];
    const float* W1  = (const float*)d_in[3];
    const float* b1  = (const float*)d_in[4];
    const float* W2  = (const float*)d_in[5];
    const float* b2  = (const float*)d_in[6];
    const float* hW1 = (const float*)d_in[7];
    const float* hb1 = (const float*)d_in[8];
    const float* hW2 = (const float*)d_in[9];
    const float* hb2 = (const float*)d_in[10];
    const float* Wc  = (const float*)d_in[11];
    const float* bc  = (const float*)d_in[12];

    const int n = in_sizes[0] / HID;   // 50000
    const int e = in_sizes[1] / 2;     // 800000
    const int* src = ei;
    const int* dst = ei + e;

    // Workspace layout (floats): dinv[n] | norm[e] | buf0[n*128] | buf1[n*128]
    float* ws   = (float*)d_ws;
    float* dinv = ws;
    float* norm = ws + n;
    float* buf0 = norm + e;
    float* buf1 = buf0 + (size_t)n * HID;

    // Output layout: z_gcn | z_hgcn | z | logits
    float* z_gcn  = (float*)d_out;
    float* z_hgcn = z_gcn + (size_t)n * HID;
    float* z      = z_hgcn + (size_t)n * HID;
    float* logits = z + (size_t)n * HID;

    const int T = 256;

    // gcn_norm
    init_deg_kernel<<<(n + T - 1) / T, T, 0, stream>>>(dinv, n);
    accum_deg_kernel<<<(e + T - 1) / T, T, 0, stream>>>(dinv, dst, ew, e);
    finalize_dinv_kernel<<<(n + T - 1) / T, T, 0, stream>>>(dinv, n);
    edge_norm_kernel<<<(e + T - 1) / T, T, 0, stream>>>(norm, dinv, src, dst, ew, e);

    const int gemmGrid = (n + 15) / 16;
    const long long msgThreads = (long long)e * HID;
    const int edgeGrid = (int)((msgThreads + T - 1) / T);
    const int initGrid = (n * 32 + T - 1) / T;
    const int nodeGrid = (n + 7) / 8;

    const float* Wenc[2][2] = {{W1, W2}, {hW1, hW2}};
    const float* benc[2][2] = {{b1, b2}, {hb1, hb2}};
    float* zenc[2] = {z_gcn, z_hgcn};

    for (int enc = 0; enc < 2; ++enc) {
        // layer 1: xw = x @ W ; acc = b + dinv^2*xw ; acc[dst] += xw[src]*norm
        gemm_k128_wmma<HID, false, false><<<gemmGrid, 256, 0, stream>>>(x, Wenc[enc][0], nullptr, buf0, n);
        scatter_init_kernel<<<initGrid, T, 0, stream>>>(buf1, buf0, dinv, benc[enc][0], n);
        scatter_edge_kernel<<<edgeGrid, T, 0, stream>>>(buf1, buf0, src, dst, norm, e);
        // layer 2: xw = relu(h) @ W  (ReLU fused into GEMM A staging)
        gemm_k128_wmma<HID, true, false><<<gemmGrid, 256, 0, stream>>>(buf1, Wenc[enc][1], nullptr, buf0, n);
        scatter_init_kernel<<<initGrid, T, 0, stream>>>(buf1, buf0, dinv, benc[enc][1], n);
        scatter_edge_kernel<<<edgeGrid, T, 0, stream>>>(buf1, buf0, src, dst, norm, e);
        // z_enc = l2norm(h2)
        l2norm_kernel<<<nodeGrid, 256, 0, stream>>>(zenc[enc], buf1, n);
    }

    // z = l2norm(0.6*z_gcn + 0.4*z_hgcn)
    combine_kernel<<<nodeGrid, 256, 0, stream>>>(z, z_gcn, z_hgcn, n);

    // logits = z @ Wc + bc : 40 cols -> 3 column tiles -> 3 waves (96 threads)
    gemm_k128_wmma<40, false, true><<<gemmGrid, 96, 0, stream>>>(z, Wc, bc, logits, n);
}